// CrfRnn3D_phlcpp_39118562132367
// MI455X (gfx1250) — compile-verified
//
#include <hip/hip_runtime.h>
#include <math.h>

#define LBL 8
#define DIM 96
#define NVOX (96*96*96)          // 884736
#define GD0 14
#define GD1 14
#define GD2 14
#define GD3 10
#define GSIZE (14*14*14*10)      // 27440
#define ST0 1960
#define ST1 140
#define ST2 10
#define INV_ALPHA 0.125f
#define INV_BETA  8.0f
#define GAMMA 3.0f
#define RAD 9

typedef __attribute__((ext_vector_type(2))) float v2f;
typedef __attribute__((ext_vector_type(8))) float v8f;

// ---------------------------------------------------------------------------
// init: Acomb[16x16] = [C@Wsp | C@Wbi ; 0], gw[19] = normalized gaussian taps
// ---------------------------------------------------------------------------
__global__ void k_init(const float* __restrict__ Wsp, const float* __restrict__ Wbi,
                       const float* __restrict__ Cm, float* __restrict__ Acomb,
                       float* __restrict__ gw) {
    int t = threadIdx.x;
    if (t < 256) {
        int m = t >> 4, k = t & 15;
        float s = 0.f;
        if (m < 8) {
            const float* W = (k < 8) ? Wsp : Wbi;
            int kk = k & 7;
            #pragma unroll
            for (int j = 0; j < 8; ++j) s += Cm[m*8 + j] * W[j*8 + kk];
        }
        Acomb[t] = s;
    }
    if (t == 0) {
        float w[2*RAD+1]; float sum = 0.f;
        for (int i = 0; i < 2*RAD+1; ++i) {
            float d = (float)(i - RAD) / GAMMA;
            w[i] = expf(-0.5f * d * d);
            sum += w[i];
        }
        for (int i = 0; i < 2*RAD+1; ++i) gw[i] = w[i] / sum;
    }
}

__global__ void k_zero(float* __restrict__ p, int n) {
    int i = blockIdx.x * blockDim.x + threadIdx.x;
    if (i < n) p[i] = 0.f;
}

// ---------------------------------------------------------------------------
// per-voxel softmax over the 8 labels: Q = softmax(U, axis=labels)
// ---------------------------------------------------------------------------
__global__ void k_softmax(const float* __restrict__ U, float* __restrict__ Q) {
    int v = blockIdx.x * blockDim.x + threadIdx.x;
    if (v >= NVOX) return;
    float u[LBL], m = -3.4e38f;
    #pragma unroll
    for (int c = 0; c < LBL; ++c) { u[c] = U[c*NVOX + v]; m = fmaxf(m, u[c]); }
    float s = 0.f;
    #pragma unroll
    for (int c = 0; c < LBL; ++c) { u[c] = expf(u[c] - m); s += u[c]; }
    float inv = 1.f / s;
    #pragma unroll
    for (int c = 0; c < LBL; ++c) Q[c*NVOX + v] = u[c] * inv;
}

// ---------------------------------------------------------------------------
// separable 19-tap gaussian along one voxel axis (zero padding)
// ---------------------------------------------------------------------------
__global__ void k_blur(const float* __restrict__ in, float* __restrict__ out,
                       const float* __restrict__ gw, int stride) {
    int t = blockIdx.x * blockDim.x + threadIdx.x;
    if (t >= LBL * NVOX) return;
    int v = t % NVOX;
    int pos = (v / stride) % DIM;
    float acc = 0.f;
    #pragma unroll
    for (int dt = -RAD; dt <= RAD; ++dt) {
        int p = pos + dt;
        if (p >= 0 && p < DIM) acc += gw[dt + RAD] * in[t + dt * stride];
    }
    out[t] = acc;
}

// ---------------------------------------------------------------------------
// splat: Qsm (+ homogeneous ones) -> 4-D bilateral grid, multilinear, atomics
// ---------------------------------------------------------------------------
__global__ void k_splat(const float* __restrict__ Q, const float* __restrict__ img,
                        float* __restrict__ grid) {
    int v = blockIdx.x * blockDim.x + threadIdx.x;
    if (v >= NVOX) return;
    int x = v % DIM, y = (v / DIM) % DIM, z = v / (DIM*DIM);
    float f0 = z * INV_ALPHA, f1 = y * INV_ALPHA, f2 = x * INV_ALPHA, f3 = img[v] * INV_BETA;
    float l0 = fminf(fmaxf(floorf(f0), 0.f), (float)(GD0-2));
    float l1 = fminf(fmaxf(floorf(f1), 0.f), (float)(GD1-2));
    float l2 = fminf(fmaxf(floorf(f2), 0.f), (float)(GD2-2));
    float l3 = fminf(fmaxf(floorf(f3), 0.f), (float)(GD3-2));
    float fr0 = f0 - l0, fr1 = f1 - l1, fr2 = f2 - l2, fr3 = f3 - l3;
    int base = (int)l0 * ST0 + (int)l1 * ST1 + (int)l2 * ST2 + (int)l3;
    float q[9];
    #pragma unroll
    for (int c = 0; c < LBL; ++c) q[c] = Q[c*NVOX + v];
    q[8] = 1.f;
    #pragma unroll
    for (int bits = 0; bits < 16; ++bits) {
        int o0 = bits & 1, o1 = (bits >> 1) & 1, o2 = (bits >> 2) & 1, o3 = (bits >> 3) & 1;
        float w = (o0 ? fr0 : 1.f - fr0) * (o1 ? fr1 : 1.f - fr1)
                * (o2 ? fr2 : 1.f - fr2) * (o3 ? fr3 : 1.f - fr3);
        int idx = base + o0*ST0 + o1*ST1 + o2*ST2 + o3;
        #pragma unroll
        for (int c = 0; c < 9; ++c) atomicAdd(&grid[c*GSIZE + idx], q[c] * w);
    }
}

// ---------------------------------------------------------------------------
// [1,2,1]/4 blur along one grid axis (zero padding)
// ---------------------------------------------------------------------------
__global__ void k_gblur(const float* __restrict__ in, float* __restrict__ out,
                        int stride, int dim) {
    int t = blockIdx.x * blockDim.x + threadIdx.x;
    if (t >= 9 * GSIZE) return;
    int g = t % GSIZE;
    int pos = (g / stride) % dim;
    float a = 0.5f * in[t];
    if (pos > 0)       a += 0.25f * in[t - stride];
    if (pos < dim - 1) a += 0.25f * in[t + stride];
    out[t] = a;
}

// ---------------------------------------------------------------------------
// fused: slice bilateral grid, 8x16 @ 16xN mix via V_WMMA_F32_16X16X4_F32,
// add unary, final softmax.  One wave handles 16 voxels; both half-waves
// redundantly compute the same voxel's X column so each lane can source its
// A/B fragments per the 32-bit 16x4 / 4x16 VGPR layouts (K split {0,1}/{2,3}).
// ---------------------------------------------------------------------------
__global__ void __launch_bounds__(256) k_fused(
        const float* __restrict__ U, const float* __restrict__ img,
        const float* __restrict__ sp, const float* __restrict__ grid,
        const float* __restrict__ Acomb, float* __restrict__ out) {
    int lane = threadIdx.x & 31;
    int wave = blockIdx.x * (blockDim.x >> 5) + (threadIdx.x >> 5);
    int nn   = lane & 15;
    int hi   = lane >> 4;              // 0: K={0,1}, 1: K={2,3} within each K4 block
    int v    = wave * 16 + nn;         // voxel (grid sized so always in range)

    // ---- bilateral slice for voxel v (9 channels) ----
    int x = v % DIM, y = (v / DIM) % DIM, z = v / (DIM*DIM);
    float f0 = z * INV_ALPHA, f1 = y * INV_ALPHA, f2 = x * INV_ALPHA, f3 = img[v] * INV_BETA;
    float l0 = fminf(fmaxf(floorf(f0), 0.f), (float)(GD0-2));
    float l1 = fminf(fmaxf(floorf(f1), 0.f), (float)(GD1-2));
    float l2 = fminf(fmaxf(floorf(f2), 0.f), (float)(GD2-2));
    float l3 = fminf(fmaxf(floorf(f3), 0.f), (float)(GD3-2));
    float fr0 = f0 - l0, fr1 = f1 - l1, fr2 = f2 - l2, fr3 = f3 - l3;
    int base = (int)l0 * ST0 + (int)l1 * ST1 + (int)l2 * ST2 + (int)l3;

    float sl[9];
    #pragma unroll
    for (int c = 0; c < 9; ++c) sl[c] = 0.f;
    #pragma unroll
    for (int bits = 0; bits < 16; ++bits) {
        int o0 = bits & 1, o1 = (bits >> 1) & 1, o2 = (bits >> 2) & 1, o3 = (bits >> 3) & 1;
        float w = (o0 ? fr0 : 1.f - fr0) * (o1 ? fr1 : 1.f - fr1)
                * (o2 ? fr2 : 1.f - fr2) * (o3 ? fr3 : 1.f - fr3);
        int idx = base + o0*ST0 + o1*ST1 + o2*ST2 + o3;
        #pragma unroll
        for (int c = 0; c < 9; ++c) sl[c] += w * grid[c*GSIZE + idx];
    }

    // ---- X column for this voxel: rows 0-7 = sp, rows 8-15 = bi ----
    float xv[16];
    float inv_den = 1.f / (sl[8] + 1e-8f);
    #pragma unroll
    for (int k = 0; k < LBL; ++k) xv[k] = sp[k*NVOX + v];
    #pragma unroll
    for (int k = 0; k < LBL; ++k) xv[8 + k] = sl[k] * inv_den;

    // ---- D(16x16) = Acomb(16x16) @ X(16x16) via 4 accumulating K=4 WMMAs ----
    v8f acc = {};
    const float* arow = Acomb + (nn << 4);
    #pragma unroll
    for (int j = 0; j < 4; ++j) {
        int kb = 4*j + 2*hi;           // this lane's two K slots in block j
        v2f a, b;
        a.x = arow[kb];
        a.y = arow[kb + 1];
        float bx0 = xv[4*j + 0], by0 = xv[4*j + 1];
        float bx1 = xv[4*j + 2], by1 = xv[4*j + 3];
        b.x = hi ? bx1 : bx0;
        b.y = hi ? by1 : by0;
        acc = __builtin_amdgcn_wmma_f32_16x16x4_f32(
                  false, a, false, b, (short)0, acc, false, false);
    }

    // ---- lanes 0-15 hold D rows 0-7 (labels) for voxel v: Q = U + Qp, softmax
    if (!hi) {
        float q[LBL], m = -3.4e38f;
        #pragma unroll
        for (int c = 0; c < LBL; ++c) { q[c] = U[c*NVOX + v] + acc[c]; m = fmaxf(m, q[c]); }
        float s = 0.f;
        #pragma unroll
        for (int c = 0; c < LBL; ++c) { q[c] = expf(q[c] - m); s += q[c]; }
        float inv = 1.f / s;
        #pragma unroll
        for (int c = 0; c < LBL; ++c) out[c*NVOX + v] = q[c] * inv;
    }
}

// ---------------------------------------------------------------------------
extern "C" void kernel_launch(void* const* d_in, const int* in_sizes, int n_in,
                              void* d_out, int out_size, void* d_ws, size_t ws_size,
                              hipStream_t stream) {
    const float* U   = (const float*)d_in[0];
    const float* img = (const float*)d_in[1];
    const float* Wsp = (const float*)d_in[2];
    const float* Wbi = (const float*)d_in[3];
    const float* Cm  = (const float*)d_in[4];
    float* out = (float*)d_out;

    float* ws    = (float*)d_ws;
    float* Acomb = ws;                     // 256
    float* gw    = ws + 256;               // 32 (19 used)
    float* gridA = ws + 512;               // 9*GSIZE
    float* gridB = gridA + 9*GSIZE;        // 9*GSIZE
    float* Qsm   = gridB + 9*GSIZE;        // LBL*NVOX
    float* T     = Qsm + LBL*NVOX;         // LBL*NVOX  (ends up holding sp)

    k_init<<<1, 256, 0, stream>>>(Wsp, Wbi, Cm, Acomb, gw);
    {
        int n = 2 * 9 * GSIZE;
        k_zero<<<(n + 255)/256, 256, 0, stream>>>(gridA, n);
    }
    k_softmax<<<(NVOX + 255)/256, 256, 0, stream>>>(U, Qsm);
    k_splat<<<(NVOX + 255)/256, 256, 0, stream>>>(Qsm, img, gridA);

    int bt = (LBL*NVOX + 255)/256;
    k_blur<<<bt, 256, 0, stream>>>(Qsm, T, gw, 1);         // x
    k_blur<<<bt, 256, 0, stream>>>(T, Qsm, gw, DIM);       // y
    k_blur<<<bt, 256, 0, stream>>>(Qsm, T, gw, DIM*DIM);   // z  -> T = sp

    int gb = (9*GSIZE + 255)/256;
    k_gblur<<<gb, 256, 0, stream>>>(gridA, gridB, ST0, GD0);
    k_gblur<<<gb, 256, 0, stream>>>(gridB, gridA, ST1, GD1);
    k_gblur<<<gb, 256, 0, stream>>>(gridA, gridB, ST2, GD2);
    k_gblur<<<gb, 256, 0, stream>>>(gridB, gridA, 1,   GD3);  // final -> gridA

    // 55296 voxel tiles of 16, 8 waves (tiles) per 256-thread block
    k_fused<<<NVOX/16/8, 256, 0, stream>>>(U, img, T, gridA, Acomb, out);
}